// NeighborlistVerletNsq_56521769615744
// MI455X (gfx1250) — compile-verified
//
#include <hip/hip_runtime.h>

// NeighborlistVerletNsq for MI455X (gfx1250, wave32).
// Store-bandwidth-bound: ~940 MB of output, ~40us floor at 23.3 TB/s.
// Strategy:
//   * grid = (N/256 j-chunks, N rows). Pair index p computed analytically from
//     (i, j) -- the 268 MB i_pairs/j_pairs inputs are never read.
//   * j-chunk positions (3 KB) staged into LDS via CDNA5 async-to-LDS loads
//     (ASYNCcnt), waited with s_wait_asynccnt 0.
//   * all outputs written with non-temporal stores (write-once stream, 5x L2).

#define BLOCK 256
#define CUTOFF 0.5f

__global__ __launch_bounds__(BLOCK) void nsq_neighbor_kernel(
    const float* __restrict__ pos,        // [N,3]
    const float* __restrict__ box,        // [3,3]
    const int*   __restrict__ is_periodic_p,
    float*       __restrict__ out,        // [7P] : idx_i | idx_j | d | r(3) | mask
    int n, long long P)
{
    const int i  = (int)blockIdx.y;
    const int jb = (int)blockIdx.x * BLOCK;
    const int t  = (int)threadIdx.x;

    // Whole chunk at or below the diagonal: no valid pair (uniform exit).
    if (jb + (BLOCK - 1) <= i) return;

    __shared__ float sPos[3 * BLOCK];

    // Async-copy this chunk's positions (3*BLOCK floats, contiguous) into LDS.
    {
        const float* gsrc = pos + (long long)jb * 3;
        const int lastE = 3 * n - 3 * jb - 1;   // clamp (n may not divide BLOCK)
#pragma unroll
        for (int k = 0; k < 3; ++k) {
            int e  = t + k * BLOCK;
            int ge = (e <= lastE) ? e : lastE;
            __builtin_amdgcn_global_load_async_to_lds_b32(
                (int*)(gsrc + ge), (int*)(&sPos[e]), 0, 0);
        }
    }

    // Uniform work while the async copy is in flight.
    const float Lx = box[0], Ly = box[4], Lz = box[8];      // diagonal box
    const float iLx = 1.0f / Lx, iLy = 1.0f / Ly, iLz = 1.0f / Lz;
    const int periodic = *is_periodic_p;
    const float xi = pos[(long long)i * 3 + 0];
    const float yi = pos[(long long)i * 3 + 1];
    const float zi = pos[(long long)i * 3 + 2];
    // Row base offset in triu(k=1) ordering.
    const long long basei =
        (long long)i * (n - 1) - ((long long)i * (i - 1)) / 2;

    asm volatile("s_wait_asynccnt 0" ::: "memory");
    __syncthreads();

    const int j = jb + t;
    if (j <= i || j >= n) return;

    float dx = xi - sPos[t * 3 + 0];
    float dy = yi - sPos[t * 3 + 1];
    float dz = zi - sPos[t * 3 + 2];

    if (periodic) {
        // jnp.remainder semantics: w - floor(w/L)*L  (result in [0, L))
        const float hx = 0.5f * Lx, hy = 0.5f * Ly, hz = 0.5f * Lz;
        float wx = dx + hx; wx -= floorf(wx * iLx) * Lx; dx = wx - hx;
        float wy = dy + hy; wy -= floorf(wy * iLy) * Ly; dy = wy - hy;
        float wz = dz + hz; wz -= floorf(wz * iLz) * Lz; dz = wz - hz;
    }

    const float d   = sqrtf(dx * dx + dy * dy + dz * dz);
    const float sel = (d <= CUTOFF) ? 1.0f : 0.0f;

    const long long p = basei + (long long)(j - i - 1);

    // pair_indices [2,P] (as float -- exact for idx < 2^24)
    __builtin_nontemporal_store((float)i, out + p);
    __builtin_nontemporal_store((float)j, out + P + p);
    // d_out [P,1] (masked)
    __builtin_nontemporal_store(sel * d, out + 2 * P + p);
    // r_out [P,3] (masked)
    float* r = out + 3 * P + 3 * p;
    __builtin_nontemporal_store(sel * dx, r + 0);
    __builtin_nontemporal_store(sel * dy, r + 1);
    __builtin_nontemporal_store(sel * dz, r + 2);
    // mask [P] as 0.0/1.0
    __builtin_nontemporal_store(sel, out + 6 * P + p);
}

extern "C" void kernel_launch(void* const* d_in, const int* in_sizes, int n_in,
                              void* d_out, int out_size, void* d_ws, size_t ws_size,
                              hipStream_t stream) {
    const float* pos = (const float*)d_in[0];   // [N,3] f32
    const float* box = (const float*)d_in[1];   // [3,3] f32
    // d_in[2] = i_pairs, d_in[3] = j_pairs : intentionally unused (computed)
    const int* isp = (const int*)d_in[4];       // scalar int
    float* out = (float*)d_out;

    const int n = in_sizes[0] / 3;
    const long long P = (long long)n * (n - 1) / 2;
    const int chunks = (n + BLOCK - 1) / BLOCK;

    dim3 grid((unsigned)chunks, (unsigned)n);
    nsq_neighbor_kernel<<<grid, BLOCK, 0, stream>>>(pos, box, isp, out, n, P);
}